// GNNInvariant_40243843563892
// MI455X (gfx1250) — compile-verified
//
#include <hip/hip_runtime.h>

// ---------------------------------------------------------------------------
// GNN message passing on MI455X (gfx1250).
// STATE_DIM=64, EDGE_DIM=1, OUTPUT_DIM=2, NUM_ROUNDS=3.
// Edge GEMM (E,65)@(65,64) done in fp32 via V_WMMA_F32_16X16X4_F32:
//   - K=0..63 contracted with 16 WMMA k-steps (K=4 each) x 4 N-tiles
//   - K=64 (edge length) folded in as a rank-1 VALU update in the epilogue
// ---------------------------------------------------------------------------

typedef float v2f __attribute__((ext_vector_type(2)));
typedef float v8f __attribute__((ext_vector_type(8)));

#define STATE_DIM 64
#define NUM_ROUNDS 3
#define TILES_PER_BLOCK 4   // 4 waves/block, one 16-edge tile per wave

__global__ void gnn_zero_kernel(float* __restrict__ p, int n) {
    int i = blockIdx.x * blockDim.x + threadIdx.x;
    if (i < n) p[i] = 0.0f;
}

__global__ void gnn_init_out_kernel(float* __restrict__ out,
                                    const float* __restrict__ b_out, int n2) {
    int i = blockIdx.x * blockDim.x + threadIdx.x;
    if (i < n2) out[i] = b_out[i & 1];
}

// One wave computes messages for 16 edges: D(16x64) = tanh(A(16x65) @ W(65x64) + b)
// and scatter-adds D into state_next with f32 atomics.
__global__ __launch_bounds__(TILES_PER_BLOCK * 32)
void gnn_msg_kernel(const float* __restrict__ state_cur,
                    float* __restrict__ state_next,
                    const float* __restrict__ edge_len,
                    const int* __restrict__ node_from,
                    const int* __restrict__ node_to,
                    const float* __restrict__ W_msg,   // (65,64) row-major
                    const float* __restrict__ b_msg,   // (64,)
                    int E) {
    // W_msg[k<64][n] pre-swizzled into the B-matrix lane layout:
    // for (ks, nt): lane L holds B[K = ks*4 + 2*(L>>4) + v][N = nt*16 + (L&15)]
    // at lds_b[ks*256 + nt*64 + L*2 + v]  -> one ds_load_b64 per WMMA.
    __shared__ float lds_b[16 * 4 * 32 * 2];   // 4096 floats = 16 KB
    __shared__ float lds_bias[STATE_DIM];
    __shared__ float lds_wlast[STATE_DIM];     // W_msg row 64 (edge feature)

    const int tid = threadIdx.x;
    for (int idx = tid; idx < 4096; idx += TILES_PER_BLOCK * 32) {
        int v    = idx & 1;
        int lane = (idx >> 1) & 31;
        int nt   = (idx >> 6) & 3;
        int ks   = idx >> 8;
        int k = ks * 4 + 2 * (lane >> 4) + v;
        int n = nt * 16 + (lane & 15);
        lds_b[idx] = W_msg[k * STATE_DIM + n];
    }
    if (tid < STATE_DIM) {
        lds_bias[tid]  = b_msg[tid];
        lds_wlast[tid] = W_msg[STATE_DIM * STATE_DIM + tid];
    }
    __syncthreads();

    const int lane = tid & 31;
    const int wave = tid >> 5;
    const int tile = blockIdx.x * TILES_PER_BLOCK + wave;
    const int ntile_total = (E + 15) >> 4;
    if (tile >= ntile_total) return;           // wave-uniform: EXEC stays all-ones

    const int e0 = tile << 4;
    const int m  = lane & 15;                  // A-matrix row this lane feeds
    const int h  = lane >> 4;                  // half selector (K pair)

    int eM = e0 + m;  if (eM > E - 1) eM = E - 1;   // clamp for partial tile
    const int src = node_from[eM];
    const float* __restrict__ arow = state_cur + (long long)src * STATE_DIM;

    v8f acc[4];
    #pragma unroll
    for (int nt = 0; nt < 4; ++nt) acc[nt] = v8f{0.f,0.f,0.f,0.f,0.f,0.f,0.f,0.f};

    // 16 k-steps x 4 n-tiles = 64 x v_wmma_f32_16x16x4_f32 per tile
    #pragma unroll
    for (int ks = 0; ks < 16; ++ks) {
        // A lane layout (ISA 7.12.2): lane L -> A[M=L&15][K = ks*4 + 2*(L>>4) + {0,1}]
        v2f a = *(const v2f*)(arow + ks * 4 + 2 * h);
        const float* bb = &lds_b[ks * 256 + lane * 2];
        #pragma unroll
        for (int nt = 0; nt < 4; ++nt) {
            v2f b = *(const v2f*)(bb + nt * 64);
            acc[nt] = __builtin_amdgcn_wmma_f32_16x16x4_f32(
                false, a, false, b, (short)0, acc[nt], false, false);
        }
    }

    // Epilogue: + edge_len * W_msg[64][n] + bias, tanh, scatter-add.
    // D layout: lane L, vgpr j -> D[M = j + 8*(L>>4)][N = L&15 (+16*nt)]
    float elen[8];
    int   dstn[8];
    bool  valid[8];
    #pragma unroll
    for (int j = 0; j < 8; ++j) {
        int e = e0 + j + 8 * h;
        valid[j] = (e < E);
        int ec = valid[j] ? e : (E - 1);
        elen[j] = edge_len[ec];
        dstn[j] = node_to[ec];
    }
    const int n = lane & 15;
    #pragma unroll
    for (int nt = 0; nt < 4; ++nt) {
        const int   col  = nt * 16 + n;
        const float bias = lds_bias[col];
        const float wl   = lds_wlast[col];
        #pragma unroll
        for (int j = 0; j < 8; ++j) {
            float val = tanhf(acc[nt][j] + elen[j] * wl + bias);
            if (valid[j])
                unsafeAtomicAdd(&state_next[(long long)dstn[j] * STATE_DIM + col], val);
        }
    }
}

// One wave per node: out[g] += state[node] @ W_out  (64 -> 2), wave32 reduction.
__global__ __launch_bounds__(256)
void gnn_pool_kernel(const float* __restrict__ state,
                     const int* __restrict__ node_graph,
                     const float* __restrict__ W_out,   // (64,2) row-major
                     float* __restrict__ out, int N) {
    const int lane = threadIdx.x & 31;
    const int node = blockIdx.x * (blockDim.x >> 5) + (threadIdx.x >> 5);
    if (node >= N) return;
    const float* __restrict__ row = state + (long long)node * STATE_DIM;
    const int k0 = 2 * lane;
    float s0 = row[k0], s1 = row[k0 + 1];
    float p0 = s0 * W_out[k0 * 2 + 0] + s1 * W_out[(k0 + 1) * 2 + 0];
    float p1 = s0 * W_out[k0 * 2 + 1] + s1 * W_out[(k0 + 1) * 2 + 1];
    #pragma unroll
    for (int off = 16; off > 0; off >>= 1) {
        p0 += __shfl_xor(p0, off, 32);
        p1 += __shfl_xor(p1, off, 32);
    }
    if (lane == 0) {
        const int g = node_graph[node];
        unsafeAtomicAdd(&out[g * 2 + 0], p0);
        unsafeAtomicAdd(&out[g * 2 + 1], p1);
    }
}

extern "C" void kernel_launch(void* const* d_in, const int* in_sizes, int n_in,
                              void* d_out, int out_size, void* d_ws, size_t ws_size,
                              hipStream_t stream) {
    // setup_inputs order:
    // 0 edge_lengths (E,1) f32 | 1 W_msg (65,64) f32 | 2 b_msg (64,) f32
    // 3 W_out (64,2) f32      | 4 b_out (2,) f32
    // 5 node_from (E,) int    | 6 node_to (E,) int   | 7 node_graph_index (N,) int
    // 8 num_nodes scalar      | 9 num_graphs scalar
    const float* edge_len  = (const float*)d_in[0];
    const float* W_msg     = (const float*)d_in[1];
    const float* b_msg     = (const float*)d_in[2];
    const float* W_out     = (const float*)d_in[3];
    const float* b_out     = (const float*)d_in[4];
    const int*   node_from = (const int*)d_in[5];
    const int*   node_to   = (const int*)d_in[6];
    const int*   node_gidx = (const int*)d_in[7];

    const int E = in_sizes[5];
    const int N = in_sizes[7];

    float* bufA = (float*)d_ws;                       // (N, 64)
    float* bufB = bufA + (size_t)N * STATE_DIM;       // (N, 64)

    const int state_elems = N * STATE_DIM;
    gnn_zero_kernel<<<(state_elems + 255) / 256, 256, 0, stream>>>(bufA, state_elems);

    const int ntiles = (E + 15) / 16;
    const int msg_blocks = (ntiles + TILES_PER_BLOCK - 1) / TILES_PER_BLOCK;

    float* cur = bufA;
    float* nxt = bufB;
    for (int r = 0; r < NUM_ROUNDS; ++r) {
        // state_next starts as state (".at[].add" semantics), then scatter-add.
        hipMemcpyAsync(nxt, cur, (size_t)state_elems * sizeof(float),
                       hipMemcpyDeviceToDevice, stream);
        gnn_msg_kernel<<<msg_blocks, TILES_PER_BLOCK * 32, 0, stream>>>(
            cur, nxt, edge_len, node_from, node_to, W_msg, b_msg, E);
        float* t = cur; cur = nxt; nxt = t;
    }

    // d_out = b_out broadcast, then per-node atomic accumulation of state@W_out.
    gnn_init_out_kernel<<<(out_size + 255) / 256, 256, 0, stream>>>(
        (float*)d_out, b_out, out_size);
    const int waves_per_block = 256 / 32;
    gnn_pool_kernel<<<(N + waves_per_block - 1) / waves_per_block, 256, 0, stream>>>(
        cur, node_gidx, W_out, (float*)d_out, N);
}